// GAT_label_51049981280742
// MI455X (gfx1250) — compile-verified
//
#include <hip/hip_runtime.h>
#include <math.h>

// ---------------- problem constants (match reference) ----------------
#define NFEAT 128
#define HC    128
#define HEADS 4
#define EMB   32
#define SLOPE 0.2f

typedef float v2f __attribute__((ext_vector_type(2)));
typedef float v8f __attribute__((ext_vector_type(8)));

__device__ __forceinline__ float eluf(float x)   { return x > 0.f ? x : (expf(x) - 1.f); }
__device__ __forceinline__ float sigf(float x)   { return 1.f / (1.f + expf(-x)); }

__device__ __forceinline__ float wave_sum32(float v) {
#pragma unroll
  for (int o = 16; o > 0; o >>= 1) v += __shfl_xor(v, o, 32);
  return v;
}

// float atomic max via sign-split integer atomics (monotone under both orders)
__device__ __forceinline__ void atomicMaxF(float* a, float v) {
  if (v >= 0.f) atomicMax((int*)a, __float_as_int(v));
  else          atomicMin((unsigned int*)a, __float_as_uint(v));
}

// =====================================================================
// fp32 WMMA GEMM:  C[M,Ncols] = A[M,K] @ B[K,Np] + bias[Np]   (row-major)
// Np = padded width (multiple of 16, B/bias zero-padded), ldc = C row stride.
// Register-blocked: one wave computes a 64x16 output slab (4 M-tiles, 4
// accumulators) so each B fragment feeds 4 V_WMMA_F32_16X16X4_F32 ops —
// the K=4 shape is fetch-limited, so B reuse is the lever that matters.
// Tail group (M/16 not divisible by 4) clamps A rows (wave-uniform) and
// guards stores; EXEC stays all-ones for every WMMA.
// ACT: 0 = none, 1 = sigmoid (fused output epilogue).
// =====================================================================
template <int ACT>
__global__ void wmma_gemm_bias(const float* __restrict__ A, const float* __restrict__ B,
                               const float* __restrict__ bias, float* __restrict__ C,
                               int M, int Np, int K, int ldc, int Ncols)
{
  const int MT = M >> 4, NT = Np >> 4;
  const int MG = (MT + 3) >> 2;             // groups of 4 M-tiles
  const int w = blockIdx.x * (blockDim.x >> 5) + (threadIdx.x >> 5);
  if (w >= MG * NT) return;                 // whole-wave guard
  const int g  = w / NT, tn = w % NT;
  const int tm0 = g * 4;
  const int lane = threadIdx.x & 31;
  const int hl   = lane >> 4;               // lane half (0/1)
  const int l16  = lane & 15;
  const int n = tn * 16 + l16;              // column owned by this lane (B/C frag)

  // A fragment base pointers for the 4 sub-tiles (tail rows clamped, uniform)
  const float* arow[4];
#pragma unroll
  for (int i = 0; i < 4; ++i) {
    const int tmi = (tm0 + i < MT) ? (tm0 + i) : (MT - 1);
    arow[i] = A + (size_t)(tmi * 16 + l16) * K + 2 * hl;  // K = {0,1}|{2,3} per half
  }

  v8f c[4];
  {
    const float b0 = bias[n];               // bias/B zero-padded to Np
#pragma unroll
    for (int i = 0; i < 4; ++i)
#pragma unroll
      for (int v = 0; v < 8; ++v) c[i][v] = b0;
  }

  for (int k = 0; k < K; k += 4) {
    v2f b;
    const float* bp = B + (size_t)(k + 2 * hl) * Np + n;  // B frag mirrors A K-split
    b.x = bp[0];
    b.y = bp[Np];
#pragma unroll
    for (int i = 0; i < 4; ++i) {
      v2f a;
      a.x = arow[i][k];
      a.y = arow[i][k + 1];
      c[i] = __builtin_amdgcn_wmma_f32_16x16x4_f32(false, a, false, b, (short)0, c[i],
                                                   false, false);
    }
  }

  if (n < Ncols) {
#pragma unroll
    for (int i = 0; i < 4; ++i) {
      if (tm0 + i < MT) {
#pragma unroll
        for (int v = 0; v < 8; ++v) {       // C/D: M = tm*16 + v + 8*half, N = n
          const int row = (tm0 + i) * 16 + v + 8 * hl;
          float val = c[i][v];
          if (ACT == 1) val = sigf(val);
          C[(size_t)row * ldc + n] = val;
        }
      }
    }
  }
}

// =====================================================================
// GAT edge phase (E real edges + N self loops)
// =====================================================================
__global__ void gat_init(float* __restrict__ m, float* __restrict__ z,
                         float* __restrict__ acc, int N)
{
  const int i = blockIdx.x * blockDim.x + threadIdx.x;
  if (i < N * HEADS) { m[i] = -INFINITY; z[i] = 0.f; }
  if (i < N * HC)    acc[i] = 0.f;
}

// pass 1: wave-per-edge, lane = channel in head; e_h = sum_c att*leakyrelu(xl[s]+xr[d])
// cache logits in ee[], running per-(dst,head) max in mmax[]
__global__ void gat_edge_logits(const float* __restrict__ xl, const float* __restrict__ xr,
                                const int* __restrict__ src, const int* __restrict__ dst,
                                const float* __restrict__ att,
                                float* __restrict__ ee, float* __restrict__ mmax,
                                int E, int N)
{
  const int w = blockIdx.x * (blockDim.x >> 5) + (threadIdx.x >> 5);
  const int lane = threadIdx.x & 31;
  if (w >= E + N) return;
  int s, d;
  if (w < E) { s = src[w]; d = dst[w]; } else { s = w - E; d = s; }
  const float* pl = xl + (size_t)s * HC;
  const float* pr = xr + (size_t)d * HC;
#pragma unroll
  for (int h = 0; h < HEADS; ++h) {
    const float u  = pl[h * EMB + lane] + pr[h * EMB + lane];
    const float lr = u > 0.f ? u : SLOPE * u;
    float v = att[h * EMB + lane] * lr;
    v = wave_sum32(v);
    if (lane == 0) {
      ee[(size_t)w * HEADS + h] = v;
      atomicMaxF(&mmax[(size_t)d * HEADS + h], v);
    }
  }
}

// pass 2: thread-per-edge; a = exp(e - m[dst]); z[dst] += a (ee overwritten with a)
__global__ void gat_edge_softmax(const int* __restrict__ dst,
                                 const float* __restrict__ mmax,
                                 float* __restrict__ ee, float* __restrict__ z,
                                 int E, int N)
{
  const int e = blockIdx.x * blockDim.x + threadIdx.x;
  if (e >= E + N) return;
  const int d = (e < E) ? dst[e] : (e - E);
#pragma unroll
  for (int h = 0; h < HEADS; ++h) {
    const float a = expf(ee[(size_t)e * HEADS + h] - mmax[(size_t)d * HEADS + h]);
    ee[(size_t)e * HEADS + h] = a;
    atomicAdd(&z[(size_t)d * HEADS + h], a);
  }
}

// pass 3: wave-per-edge; acc[dst] += (a / max(z,1e-16)) * xl[src]
__global__ void gat_edge_scatter(const float* __restrict__ xl,
                                 const int* __restrict__ src, const int* __restrict__ dst,
                                 const float* __restrict__ ee, const float* __restrict__ z,
                                 float* __restrict__ acc, int E, int N)
{
  const int w = blockIdx.x * (blockDim.x >> 5) + (threadIdx.x >> 5);
  const int lane = threadIdx.x & 31;
  if (w >= E + N) return;
  int s, d;
  if (w < E) { s = src[w]; d = dst[w]; } else { s = w - E; d = s; }
  const float* pl = xl + (size_t)s * HC;
#pragma unroll
  for (int h = 0; h < HEADS; ++h) {
    const float zz    = z[(size_t)d * HEADS + h];
    const float alpha = ee[(size_t)w * HEADS + h] / fmaxf(zz, 1e-16f);
    atomicAdd(&acc[(size_t)d * HC + h * EMB + lane], alpha * pl[h * EMB + lane]);
  }
}

__global__ void gat_finalize_elu(const float* __restrict__ acc, const float* __restrict__ bias,
                                 float* __restrict__ out, int total)
{
  const int i = blockIdx.x * blockDim.x + threadIdx.x;
  if (i >= total) return;
  out[i] = eluf(acc[i] + bias[i & (HC - 1)]);
}

// =====================================================================
// Label GCN (52 nodes, 416 edges) fully in LDS, single block.
// Emits zero-padded WfL[128,64] = Wf @ L_final and bfL[64] = bf @ L_final
// so the output projection folds into one WMMA GEMM + sigmoid.
// =====================================================================
__global__ void label_gcn(const float* __restrict__ label_x, const int* __restrict__ ledge,
                          const float* __restrict__ lew,
                          const float* __restrict__ Wg0, const float* __restrict__ bg0,
                          const float* __restrict__ Wg1, const float* __restrict__ bg1,
                          const float* __restrict__ Wf,  const float* __restrict__ bf,
                          float* __restrict__ WfL, float* __restrict__ bfL, int LE)
{
  __shared__ float L[52 * 52];
  __shared__ float H[52 * 64];
  __shared__ float XW[52 * 64];
  __shared__ float dinv[52];
  __shared__ float ssum;
  const int tid = threadIdx.x;      // 256 threads
  const int* lsrc = ledge;
  const int* ldst = ledge + LE;

  if (tid == 0) { float s = 0.f; for (int i = 0; i < 52; ++i) s += label_x[i]; ssum = s; }
  if (tid < 52) dinv[tid] = 1.0f;   // deg accumulator, self-loop weight 1
  __syncthreads();
  for (int i = tid; i < 52 * 52; i += 256) {
    const int r = i / 52, cc = i % 52;
    L[i] = (r == cc) ? label_x[r] / ssum : 0.f;
  }
  for (int e = tid; e < LE; e += 256) atomicAdd(&dinv[ldst[e]], lew[e]);
  __syncthreads();
  if (tid < 52) { const float dg = dinv[tid]; dinv[tid] = dg > 0.f ? rsqrtf(dg) : 0.f; }
  __syncthreads();

  // ---- layer 0: XW = L @ Wg0 [52,64]; aggregate; elu ----
  for (int i = tid; i < 52 * 64; i += 256) {
    const int r = i / 64, f = i % 64;
    float s = 0.f;
    for (int k = 0; k < 52; ++k) s += L[r * 52 + k] * Wg0[k * 64 + f];
    XW[i] = s;
  }
  __syncthreads();
  for (int i = tid; i < 52 * 64; i += 256) { const int r = i / 64; H[i] = dinv[r] * dinv[r] * XW[i]; }
  __syncthreads();
  for (int idx = tid; idx < LE * 64; idx += 256) {
    const int e = idx / 64, f = idx % 64;
    const int s = lsrc[e], d = ldst[e];
    atomicAdd(&H[d * 64 + f], dinv[s] * lew[e] * dinv[d] * XW[s * 64 + f]);
  }
  __syncthreads();
  for (int i = tid; i < 52 * 64; i += 256) H[i] = eluf(H[i] + bg0[i % 64]);
  __syncthreads();

  // ---- layer 1: XW = H @ Wg1 [52,52]; aggregate into L; elu ----
  for (int i = tid; i < 52 * 52; i += 256) {
    const int r = i / 52, cc = i % 52;
    float s = 0.f;
    for (int k = 0; k < 64; ++k) s += H[r * 64 + k] * Wg1[k * 52 + cc];
    XW[i] = s;
  }
  __syncthreads();
  for (int i = tid; i < 52 * 52; i += 256) { const int r = i / 52; L[i] = dinv[r] * dinv[r] * XW[i]; }
  __syncthreads();
  for (int idx = tid; idx < LE * 52; idx += 256) {
    const int e = idx / 52, cc = idx % 52;
    const int s = lsrc[e], d = ldst[e];
    atomicAdd(&L[d * 52 + cc], dinv[s] * lew[e] * dinv[d] * XW[s * 52 + cc]);
  }
  __syncthreads();
  for (int i = tid; i < 52 * 52; i += 256) L[i] = eluf(L[i] + bg1[i % 52]);
  __syncthreads();

  // ---- fold: WfL = Wf @ L (pad to 64 cols with zeros), bfL = bf @ L ----
  for (int i = tid; i < 128 * 64; i += 256) {
    const int r = i / 64, cc = i % 64;
    float s = 0.f;
    if (cc < 52) for (int k = 0; k < 52; ++k) s += Wf[r * 52 + k] * L[k * 52 + cc];
    WfL[i] = s;
  }
  if (tid < 64) {
    float s = 0.f;
    if (tid < 52) for (int k = 0; k < 52; ++k) s += bf[k] * L[k * 52 + tid];
    bfL[tid] = s;
  }
}

// =====================================================================
// Host-side orchestration
// =====================================================================
extern "C" void kernel_launch(void* const* d_in, const int* in_sizes, int n_in,
                              void* d_out, int out_size, void* d_ws, size_t ws_size,
                              hipStream_t stream)
{
  const float* x    = (const float*)d_in[0];
  const int*   ei   = (const int*)  d_in[1];
  const float* lx   = (const float*)d_in[2];
  const int*   lei  = (const int*)  d_in[3];
  const float* lew  = (const float*)d_in[4];
  const float* Wl0  = (const float*)d_in[5];  const float* bl0 = (const float*)d_in[6];
  const float* Wr0  = (const float*)d_in[7];  const float* br0 = (const float*)d_in[8];
  const float* att0 = (const float*)d_in[9];  const float* b0  = (const float*)d_in[10];
  const float* Wl1  = (const float*)d_in[11]; const float* bl1 = (const float*)d_in[12];
  const float* Wr1  = (const float*)d_in[13]; const float* br1 = (const float*)d_in[14];
  const float* att1 = (const float*)d_in[15]; const float* b1  = (const float*)d_in[16];
  const float* Wg0  = (const float*)d_in[17]; const float* bg0 = (const float*)d_in[18];
  const float* Wg1  = (const float*)d_in[19]; const float* bg1 = (const float*)d_in[20];
  const float* Wf   = (const float*)d_in[21]; const float* bf  = (const float*)d_in[22];

  const int Nn = in_sizes[0] / NFEAT;   // 50000
  const int E  = in_sizes[1] / 2;       // 800000
  const int LE = in_sizes[3] / 2;       // 416
  const int TE = E + Nn;                // edges incl. self loops

  const int* src = ei;
  const int* dst = ei + E;

  // workspace carve-up (floats)
  float* wsf  = (float*)d_ws;
  const size_t NB = (size_t)Nn * HC;
  float* xl   = wsf;
  float* xr   = xl  + NB;
  float* acc  = xr  + NB;
  float* hbuf = acc + NB;
  float* mbuf = hbuf + NB;
  float* zbuf = mbuf + (size_t)Nn * HEADS;
  float* ee   = zbuf + (size_t)Nn * HEADS;
  float* WfL  = ee   + (size_t)TE * HEADS;
  float* bfL  = WfL  + 128 * 64;
  (void)n_in; (void)out_size; (void)ws_size;

  const int MT = Nn / 16;                                    // 3125
  const int MG = (MT + 3) / 4;                               // 782 (64-row slabs)
  const int gemm_blocks   = (MG * (HC / 16) + 7) / 8;        // 8 waves/block
  const int gemmO_blocks  = (MG * (64 / 16) + 7) / 8;
  const int edge_wblocks  = (TE + 7) / 8;                    // wave-per-edge kernels
  const int edge_tblocks  = (TE + 255) / 256;
  const int elem_blocks   = ((int)NB + 255) / 256;

  // label GCN is independent — launch first
  label_gcn<<<1, 256, 0, stream>>>(lx, lei, lew, Wg0, bg0, Wg1, bg1, Wf, bf, WfL, bfL, LE);

  // ---------------- GAT layer 0 ----------------
  wmma_gemm_bias<0><<<gemm_blocks, 256, 0, stream>>>(x, Wl0, bl0, xl, Nn, HC, NFEAT, HC, HC);
  wmma_gemm_bias<0><<<gemm_blocks, 256, 0, stream>>>(x, Wr0, br0, xr, Nn, HC, NFEAT, HC, HC);
  gat_init<<<elem_blocks, 256, 0, stream>>>(mbuf, zbuf, acc, Nn);
  gat_edge_logits <<<edge_wblocks, 256, 0, stream>>>(xl, xr, src, dst, att0, ee, mbuf, E, Nn);
  gat_edge_softmax<<<edge_tblocks, 256, 0, stream>>>(dst, mbuf, ee, zbuf, E, Nn);
  gat_edge_scatter<<<edge_wblocks, 256, 0, stream>>>(xl, src, dst, ee, zbuf, acc, E, Nn);
  gat_finalize_elu<<<elem_blocks, 256, 0, stream>>>(acc, b0, hbuf, (int)NB);

  // ---------------- GAT layer 1 ----------------
  wmma_gemm_bias<0><<<gemm_blocks, 256, 0, stream>>>(hbuf, Wl1, bl1, xl, Nn, HC, HC, HC, HC);
  wmma_gemm_bias<0><<<gemm_blocks, 256, 0, stream>>>(hbuf, Wr1, br1, xr, Nn, HC, HC, HC, HC);
  gat_init<<<elem_blocks, 256, 0, stream>>>(mbuf, zbuf, acc, Nn);
  gat_edge_logits <<<edge_wblocks, 256, 0, stream>>>(xl, xr, src, dst, att1, ee, mbuf, E, Nn);
  gat_edge_softmax<<<edge_tblocks, 256, 0, stream>>>(dst, mbuf, ee, zbuf, E, Nn);
  gat_edge_scatter<<<edge_wblocks, 256, 0, stream>>>(xl, src, dst, ee, zbuf, acc, E, Nn);
  gat_finalize_elu<<<elem_blocks, 256, 0, stream>>>(acc, b1, hbuf, (int)NB);

  // ---------------- fused output: sigmoid(h @ (Wf@L) + bf@L) ----------------
  wmma_gemm_bias<1><<<gemmO_blocks, 256, 0, stream>>>(hbuf, WfL, bfL, (float*)d_out,
                                                      Nn, 64, HC, 52, 52);
}